// biquadratic_bspline_surface_fit_27084063768633
// MI455X (gfx1250) — compile-verified
//
#include <hip/hip_runtime.h>

// ---------------------------------------------------------------------------
// Biquadratic B-spline surface normals, 2M points, MI455X (gfx1250).
// Bandwidth/gather-bound: ~40 MB streamed + L2-resident 2 MB control grid.
// No WMMA by design: per-point math is a lane-local 3x3 contraction; matrix
// cores cannot be fed faster than the L2 gather that dominates this op.
// This revision removes the 4 precise f32 divisions (~40 VALU ops/thread,
// replaced by 1 reciprocal-multiply FMA each; C1 continuity of the quadratic
// B-spline makes any 1-ulp floor boundary flip numerically harmless) and
// processes 4 points/thread so the output becomes 3x aligned b128 NT stores.
// ---------------------------------------------------------------------------

#define GRID_D 0.14142135623730950488016887242097  // 2*sqrt(100*100/2e6)
static constexpr float X0F    = (float)(0.0 - GRID_D);
static constexpr float Y0F    = (float)(0.0 - GRID_D);
static constexpr float RGRIDF = (float)(1.0 / GRID_D);
static constexpr float INVF   = (float)(1.0 / (4.0 * GRID_D));

typedef float v4f __attribute__((ext_vector_type(4)));

__device__ __forceinline__ void eval_normal(float x, float y,
                                            const float* __restrict__ ctrl,
                                            int M,
                                            float& nx, float& ny, float& nz) {
  // Cell coordinates. tx,ty >= 1.5 always (domain [0,100] + guard ring),
  // so floor == truncate and every 3x3 gather is in-bounds.
  const float tx = fmaf(x - X0F, RGRIDF, 0.5f);
  const float ty = fmaf(y - Y0F, RGRIDF, 0.5f);
  const float fi = floorf(tx);
  const float fj = floorf(ty);
  const float u = tx - fi;
  const float w = ty - fj;
  const int ii = (int)fi;
  const int jj = (int)fj;

  // Quadratic B-spline basis + derivatives (u direction)
  const float cu  = u * u;
  const float dcu = 2.0f * u;
  const float bu  = -2.0f * cu + dcu + 1.0f;
  const float dbu = 2.0f - 4.0f * u;
  const float au  = cu - dcu + 1.0f;
  const float dau = 2.0f * u - 2.0f;
  // (w direction)
  const float cw  = w * w;
  const float dcw = 2.0f * w;
  const float bw  = -2.0f * cw + dcw + 1.0f;
  const float dbw = 2.0f - 4.0f * w;
  const float aw  = cw - dcw + 1.0f;
  const float daw = 2.0f * w - 2.0f;

  // Gather 3 rows of 3 consecutive control points (L2-resident, b96-merged)
  const float* __restrict__ row = ctrl + (long)(jj - 1) * M + (ii - 1);

  float s_du = 0.0f;  // accumulates Wb[b]  * (dU . Z[b,:])
  float s_u  = 0.0f;  // accumulates dWb[b] * ( U . Z[b,:])

  {  // b = 0  (weight aw / daw)
    const float z0 = row[0], z1 = row[1], z2 = row[2];
    const float rd = fmaf(dau, z0, fmaf(dbu, z1, dcu * z2));
    const float ru = fmaf(au,  z0, fmaf(bu,  z1, cu  * z2));
    s_du = fmaf(aw,  rd, s_du);
    s_u  = fmaf(daw, ru, s_u);
  }
  row += M;
  {  // b = 1  (weight bw / dbw)
    const float z0 = row[0], z1 = row[1], z2 = row[2];
    const float rd = fmaf(dau, z0, fmaf(dbu, z1, dcu * z2));
    const float ru = fmaf(au,  z0, fmaf(bu,  z1, cu  * z2));
    s_du = fmaf(bw,  rd, s_du);
    s_u  = fmaf(dbw, ru, s_u);
  }
  row += M;
  {  // b = 2  (weight cw / dcw)
    const float z0 = row[0], z1 = row[1], z2 = row[2];
    const float rd = fmaf(dau, z0, fmaf(dbu, z1, dcu * z2));
    const float ru = fmaf(au,  z0, fmaf(bu,  z1, cu  * z2));
    s_du = fmaf(cw,  rd, s_du);
    s_u  = fmaf(dcw, ru, s_u);
  }

  const float pzpu = s_du * INVF;
  const float pzpw = s_u * INVF;

  // Normalize (pzpu, pzpw, -1): v_rsq_f32 + one Newton step (~f32 accurate)
  const float s = fmaf(pzpu, pzpu, fmaf(pzpw, pzpw, 1.0f));
#if __has_builtin(__builtin_amdgcn_rsqf)
  float r = __builtin_amdgcn_rsqf(s);
#else
  float r = rsqrtf(s);
#endif
  r = r * fmaf(-0.5f * s * r, r, 1.5f);

  nx = pzpu * r;
  ny = pzpw * r;
  nz = -r;
}

__global__ __launch_bounds__(256) void bspline_normals_kernel(
    const float* __restrict__ xy,    // [N,2] interleaved
    const float* __restrict__ ctrl,  // [N_CTRL, M_CTRL]
    float* __restrict__ out,         // [N,3] interleaved
    int nquads,                      // N/4 (each thread does 4 points)
    int M) {                         // M_CTRL (row stride of ctrl)
  const int t = blockIdx.x * blockDim.x + threadIdx.x;
  if (t >= nquads) return;

  // Two 16B NT loads cover four points' (x,y).
  const v4f p01 = __builtin_nontemporal_load((const v4f*)xy + 2 * t);
  const v4f p23 = __builtin_nontemporal_load((const v4f*)xy + 2 * t + 1);

  float ax, ay, az, bx, by, bz, cx, cy, cz, dx, dy, dz;
  eval_normal(p01.x, p01.y, ctrl, M, ax, ay, az);
  eval_normal(p01.z, p01.w, ctrl, M, bx, by, bz);
  eval_normal(p23.x, p23.y, ctrl, M, cx, cy, cz);
  eval_normal(p23.z, p23.w, ctrl, M, dx, dy, dz);

  // 48B contiguous per lane -> three 16B-aligned nontemporal b128 stores,
  // perfectly coalesced across the wave. NT keeps the 2MB ctrl grid hot in L2.
  v4f* o = (v4f*)out + 3 * (size_t)t;
  __builtin_nontemporal_store((v4f){ax, ay, az, bx}, o + 0);
  __builtin_nontemporal_store((v4f){by, bz, cx, cy}, o + 1);
  __builtin_nontemporal_store((v4f){cz, dx, dy, dz}, o + 2);
}

extern "C" void kernel_launch(void* const* d_in, const int* in_sizes, int n_in,
                              void* d_out, int out_size, void* d_ws, size_t ws_size,
                              hipStream_t stream) {
  // Inputs in setup_inputs() order:
  //   d_in[0] = data_normal   (float32, UNUSED by the reference computation)
  //   d_in[1] = data_xy       (float32, [N,2])
  //   d_in[2] = ctrl_points_z (float32, [N_CTRL, M_CTRL])
  (void)n_in; (void)d_ws; (void)ws_size; (void)out_size;

  const float* xy   = (const float*)d_in[1];
  const float* ctrl = (const float*)d_in[2];
  float* out        = (float*)d_out;

  const int n_pts  = in_sizes[1] / 2;  // flat count / 2 coords
  const int nquads = n_pts / 4;        // 4 points per thread (N = 2e6, /4 ok)

  // M_CTRL == N_CTRL (square grid): integer sqrt of the flat ctrl size.
  int M = 1;
  while ((long)M * M < (long)in_sizes[2]) ++M;

  const int block = 256;  // 8 wave32s
  const int grid  = (nquads + block - 1) / block;
  bspline_normals_kernel<<<grid, block, 0, stream>>>(xy, ctrl, out, nquads, M);
}